// ROIPoolingLayer_76029511074133
// MI455X (gfx1250) — compile-verified
//
#include <hip/hip_runtime.h>
#include <math.h>

// ROI max-pooling for MI455X (gfx1250).
// B=4, C=128, H=50, W=50, K=128, P=7. Output (K, C, 7, 7) f32.
//
// CDNA5 path: async global->LDS staging (GLOBAL_LOAD_ASYNC_TO_LDS_B64/B32,
// ASYNCcnt + s_wait_asynccnt). WMMA is not applicable: max-pool is a
// max-reduction, not a MAC.

#define P_BINS 7
#define B_DIM  4
#define C_DIM  128
#define H_DIM  50
#define W_DIM  50
#define K_DIM  128
#define CT     4            // channels per block
#define CBLK   (C_DIM / CT) // 32 channel-tiles
#define THREADS 256
#define CHPIX  (H_DIM * W_DIM)   // 2500 floats per channel plane

#if defined(__gfx1250__) && __has_builtin(__builtin_amdgcn_global_load_async_to_lds_b32)
#define ROI_ASYNC_LDS 1
#endif

typedef int v2i_t __attribute__((vector_size(2 * sizeof(int))));
typedef __attribute__((address_space(1))) int   gint;    // global int
typedef __attribute__((address_space(3))) int   lint;    // LDS int
typedef __attribute__((address_space(1))) v2i_t gv2i;    // global int2
typedef __attribute__((address_space(3))) v2i_t lv2i;    // LDS int2

// Copy 8 bytes (two f32) global -> LDS. Caller guarantees 8B alignment.
__device__ __forceinline__ void stage_b64(const float* g, float* l) {
#ifdef ROI_ASYNC_LDS
#if __has_builtin(__builtin_amdgcn_global_load_async_to_lds_b64)
    __builtin_amdgcn_global_load_async_to_lds_b64(
        (gv2i*)g, (lv2i*)l, /*offset=*/0, /*cpol=*/0);
#else
    __builtin_amdgcn_global_load_async_to_lds_b32((gint*)g,     (lint*)l,     0, 0);
    __builtin_amdgcn_global_load_async_to_lds_b32((gint*)g + 1, (lint*)l + 1, 0, 0);
#endif
#else
    l[0] = g[0];
    l[1] = g[1];
#endif
}

__device__ __forceinline__ void stage_drain() {
#ifdef ROI_ASYNC_LDS
#if __has_builtin(__builtin_amdgcn_s_wait_asynccnt)
    __builtin_amdgcn_s_wait_asynccnt(0);
#else
    asm volatile("s_wait_asynccnt 0" ::: "memory");
#endif
#endif
}

__global__ __launch_bounds__(THREADS)
void roi_pool_gfx1250(const float* __restrict__ feat,
                      const float* __restrict__ props,
                      float* __restrict__ out)
{
    __shared__ float region[CT * CHPIX];             // 40000 B: staged rows (full width)
    __shared__ float colmax[CT * P_BINS * H_DIM];    //  5600 B: per-bin column maxima
    __shared__ int   s_wlo[P_BINS], s_whi[P_BINS];
    __shared__ int   s_hlo[P_BINS], s_hhi[P_BINS];
    __shared__ int   s_bidx;

    const int blk = blockIdx.x;
    const int k   = blk / CBLK;          // proposal index
    const int c0  = (blk % CBLK) * CT;   // first channel of this tile
    const int tid = threadIdx.x;

    // ---- bin bounds (threads 0..6), replicating the reference f32 math ----
    if (tid < P_BINS) {
        const float* p = props + (size_t)k * 5;
        const int x1 = (int)rintf(p[1]);   // SCALE == 1.0, jnp.round == rint
        const int y1 = (int)rintf(p[2]);
        const int x2 = (int)rintf(p[3]);
        const int y2 = (int)rintf(p[4]);
        const int roi_w = max(x2 - x1 + 1, 1);
        const int roi_h = max(y2 - y1 + 1, 1);
        const float sw = (float)roi_w / 7.0f;
        const float sh = (float)roi_h / 7.0f;
        const float bf = (float)tid;
        int wlo = (int)floorf(bf * sw) + x1;
        int whi = (int)ceilf((bf + 1.0f) * sw) + x1;
        int hlo = (int)floorf(bf * sh) + y1;
        int hhi = (int)ceilf((bf + 1.0f) * sh) + y1;
        s_wlo[tid] = min(max(wlo, 0), W_DIM);
        s_whi[tid] = min(max(whi, 0), W_DIM);
        s_hlo[tid] = min(max(hlo, 0), H_DIM);
        s_hhi[tid] = min(max(hhi, 0), H_DIM);
        if (tid == 0) s_bidx = (int)rintf(p[0]);
    }
    __syncthreads();

    const int bidx = s_bidx;
    const int hmin = s_hlo[0];           // lo/hi are monotone in bin index
    const int hmax = s_hhi[P_BINS - 1];
    const int nh   = hmax - hmin;        // rows needed (may be <= 0)

    // ---- async-stage rows [hmin,hmax) x [0,W) for CT channels into LDS ----
    // Per channel the span is nh*W contiguous floats (even count), 8B aligned
    // both in global (base 256B-aligned, offsets multiple of 200B/10000B) and
    // in LDS, so b64 async copies tile it exactly.
    if (nh > 0) {
        const int rowspan = nh * W_DIM;       // even
        const float* src = feat + ((size_t)bidx * C_DIM + c0) * CHPIX
                                + hmin * W_DIM;
#pragma unroll
        for (int c = 0; c < CT; ++c) {
            const float* srcc = src + c * CHPIX;
            float*       dstc = &region[c * rowspan];
            for (int r = 2 * tid; r < rowspan; r += 2 * THREADS) {
                stage_b64(srcc + r, dstc + r);
            }
        }
    }
    stage_drain();
    __syncthreads();

    // ---- phase A: column max per (c, row, pw) ----
    // Fixed CT*7*50 item space: constant divisors (no runtime div); rows
    // beyond nh are skipped.
    if (nh > 0) {
        const int rowspan = nh * W_DIM;
        for (int i = tid; i < CT * P_BINS * H_DIM; i += THREADS) {
            const int hh = i % H_DIM;             // row relative to hmin
            if (hh >= nh) continue;
            const int t  = i / H_DIM;             // c*7 + pw
            const int pw = t % P_BINS;
            const int c  = t / P_BINS;
            const int lo = s_wlo[pw], hi = s_whi[pw];
            float m = -__builtin_inff();          // empty w-bin stays -inf
            const float* rowp = &region[c * rowspan + hh * W_DIM];
            for (int w = lo; w < hi; ++w) m = fmaxf(m, rowp[w]);
            colmax[t * H_DIM + hh] = m;
        }
    }
    __syncthreads();

    // ---- phase B: row max per (c, ph, pw); empty bin -> 0 ----
    for (int i = tid; i < CT * P_BINS * P_BINS; i += THREADS) {
        const int pw = i % P_BINS;
        const int ph = (i / P_BINS) % P_BINS;
        const int c  = i / (P_BINS * P_BINS);
        const int hlo = s_hlo[ph], hhi = s_hhi[ph];
        const int wlo = s_wlo[pw], whi = s_whi[pw];
        float r;
        if ((hhi <= hlo) || (whi <= wlo)) {
            r = 0.0f;
        } else {
            float m = -__builtin_inff();
            const float* cm = &colmax[(c * P_BINS + pw) * H_DIM];
            for (int hh = hlo; hh < hhi; ++hh) m = fmaxf(m, cm[hh - hmin]);
            r = m;
        }
        out[(((size_t)k * C_DIM + (c0 + c)) * P_BINS + ph) * P_BINS + pw] = r;
    }
}

extern "C" void kernel_launch(void* const* d_in, const int* in_sizes, int n_in,
                              void* d_out, int out_size, void* d_ws, size_t ws_size,
                              hipStream_t stream) {
    (void)in_sizes; (void)n_in; (void)d_ws; (void)ws_size; (void)out_size;
    const float* feat  = (const float*)d_in[0];   // (4,128,50,50) f32
    const float* props = (const float*)d_in[1];   // (128,5) f32
    float* out = (float*)d_out;                   // (128,128,7,7) f32

    dim3 grid(K_DIM * CBLK);   // 128 proposals x 32 channel-tiles = 4096 blocks
    dim3 block(THREADS);
    roi_pool_gfx1250<<<grid, block, 0, stream>>>(feat, props, out);
}